// CapsuleLayer_75634374082759
// MI455X (gfx1250) — compile-verified
//
#include <hip/hip_runtime.h>

typedef float v2f __attribute__((ext_vector_type(2)));
typedef float v8f __attribute__((ext_vector_type(8)));

#define RNODES   1152
#define RSTRIDE  1156        // padded LDS row stride: 1156 % 64 banks == 4
#define KTOT     9216        // RNODES * 8
#define BT       16          // batches per block (WMMA M)
#define NWAVE    16
#define NTHREADS 512
#define RPW      (RNODES / NWAVE)   // 72 routes per wave
#define TSTRIDE  17                  // padded 16x16 tile row stride
#define TSIZE    (16 * TSTRIDE)      // 272 floats per tile slot

// One workgroup = (capsule c, 16-batch tile); 16 waves split the route/K dim.
// Fully fused routing: priors never materialized; recomputed inside WMMA GEMMs.
// x (9.4MB) + W (5.9MB) are L2-resident; ~15MB HBM traffic total.
__global__ __launch_bounds__(NTHREADS)
void capsule_route_kernel(const float* __restrict__ xg,   // [256][1152][8]
                          const float* __restrict__ wg,   // [10][1152][8][16]
                          float* __restrict__ out)        // [256][10][16]
{
    __shared__ float l_logits[BT * RSTRIDE];   // logits[b][r]
    __shared__ float s_probs [BT * RSTRIDE];   // exp(logit - rowmax), unnormalized
    __shared__ float s_red[NWAVE * TSIZE];     // per-wave 16x16 tiles (reduce / u)
    __shared__ float s_row[NTHREADS];          // row-reduction scratch
    __shared__ float s_v[256];                 // squashed outputs v[b][o]
    __shared__ float s_inv[BT];                // 1 / sum_r exp

    const int tid  = threadIdx.x;
    const int w    = tid >> 5;
    const int lane = tid & 31;
    const int half = lane >> 4;     // fragment half: lanes 0-15 vs 16-31
    const int bl   = lane & 15;     // M (batch) / N (col) index within fragment
    const int bt   = blockIdx.x;    // batch tile 0..15
    const int c    = blockIdx.y;    // capsule 0..9

    for (int idx = tid; idx < BT * RSTRIDE; idx += NTHREADS) l_logits[idx] = 0.0f;
    __syncthreads();

    const float* xrow   = xg + (size_t)(bt * BT + bl) * KTOT;  // x[b=bl][k]
    const float* wcbase = wg + (size_t)c * KTOT * 16;          // W_c[k][o]

    for (int iter = 0; iter < 3; ++iter) {
        // ---------- softmax over r per batch: fill s_probs, s_inv ----------
        {
            const int b = tid >> 5, slot = tid & 31;
            float m = -3.4e38f;
            for (int r = slot; r < RNODES; r += 32)
                m = fmaxf(m, l_logits[b * RSTRIDE + r]);
            s_row[tid] = m;
            __syncthreads();
            float mb = s_row[b * 32];
            for (int t = 1; t < 32; ++t) mb = fmaxf(mb, s_row[b * 32 + t]);
            __syncthreads();
            float sum = 0.0f;
            for (int r = slot; r < RNODES; r += 32) {
                const float pu = __expf(l_logits[b * RSTRIDE + r] - mb);
                s_probs[b * RSTRIDE + r] = pu;
                sum += pu;
            }
            s_row[tid] = sum;
            __syncthreads();
            if (slot == 0) {
                float s = 0.0f;
                for (int t = 0; t < 32; ++t) s += s_row[b * 32 + t];
                s_inv[b] = 1.0f / s;
            }
            __syncthreads();
        }

        // ---------- s[b,o] = sum_k (p[b,k/8]*x[b,k]) * Wc[k,o] ----------
        // A-frag (16x4 f32): lanes<16 hold K=k0,k0+1; lanes>=16 hold K=k0+2,k0+3
        {
            v8f accA = {}, accB = {};
            const int   rbeg = w * RPW;
            const float pi   = s_inv[bl];
            const float* wb  = wcbase + bl;        // + k*16 : Wc[k][o=bl]
            for (int r = rbeg; r < rbeg + RPW; ++r) {
                const float p = s_probs[bl * RSTRIDE + r] * pi;
                int k0 = r * 8 + 2 * half;
                v2f A, B;
                A.x = xrow[k0]     * p;
                A.y = xrow[k0 + 1] * p;
                B.x = wb[(size_t)k0 * 16];
                B.y = wb[(size_t)(k0 + 1) * 16];
                accA = __builtin_amdgcn_wmma_f32_16x16x4_f32(
                         false, A, false, B, (short)0, accA, false, false);
                k0 += 4;
                A.x = xrow[k0]     * p;
                A.y = xrow[k0 + 1] * p;
                B.x = wb[(size_t)k0 * 16];
                B.y = wb[(size_t)(k0 + 1) * 16];
                accB = __builtin_amdgcn_wmma_f32_16x16x4_f32(
                         false, A, false, B, (short)0, accB, false, false);
            }
            // C/D layout: VGPR v holds M=v+8*half, N=lane&15
            for (int vv = 0; vv < 8; ++vv)
                s_red[w * TSIZE + (vv + 8 * half) * TSTRIDE + bl] = accA[vv] + accB[vv];
        }
        __syncthreads();

        // ---------- cross-wave reduce + squash (first 8 waves) ----------
        if (tid < 256) {
            float sacc = 0.0f;
            for (int ww = 0; ww < NWAVE; ++ww)
                sacc += s_red[ww * TSIZE + (tid >> 4) * TSTRIDE + (tid & 15)];
            s_row[tid] = sacc * sacc;
            s_v[tid]   = sacc;          // stash; rescaled after row-sum
        }
        __syncthreads();
        if (tid < 256) {
            const int b = tid >> 4;
            float sq = 0.0f;
            for (int o = 0; o < 16; ++o) sq += s_row[b * 16 + o];
            const float scale = sqrtf(sq) / (1.0f + sq);  // = sq/((1+sq)*sqrt(sq))
            const float vo = s_v[tid] * scale;
            s_v[tid] = vo;
            if (iter == 2)
                out[(size_t)(bt * BT + b) * 160 + c * 16 + (tid & 15)] = vo;
        }
        __syncthreads();

        // ---------- delta logits (iters 0,1) ----------
        // u[b,k] = sum_o v[b,o]*Wc[k,o] via WMMA (K=16 over o, 16 k's per chunk),
        // then delta[b,r] = sum_i x[b,r*8+i]*u[b,r*8+i]
        if (iter < 2) {
            float* uslot = s_red + w * TSIZE;
            for (int j = 0; j < RPW / 2; ++j) {
                const int kb = w * RPW * 8 + j * 16;
                v8f a0 = {}, a1 = {};
                const float* wr = wcbase + (size_t)(kb + bl) * 16 + 2 * half;
                for (int ob = 0; ob < 16; ob += 8) {
                    v2f A, B;
                    A.x = s_v[bl * 16 + ob + 2 * half];
                    A.y = s_v[bl * 16 + ob + 2 * half + 1];
                    B.x = wr[ob];
                    B.y = wr[ob + 1];
                    a0 = __builtin_amdgcn_wmma_f32_16x16x4_f32(
                           false, A, false, B, (short)0, a0, false, false);
                    A.x = s_v[bl * 16 + ob + 4 + 2 * half];
                    A.y = s_v[bl * 16 + ob + 5 + 2 * half];
                    B.x = wr[ob + 4];
                    B.y = wr[ob + 5];
                    a1 = __builtin_amdgcn_wmma_f32_16x16x4_f32(
                           false, A, false, B, (short)0, a1, false, false);
                }
                for (int vv = 0; vv < 8; ++vv)
                    uslot[(vv + 8 * half) * TSTRIDE + bl] = a0[vv] + a1[vv];
                // same-wave LDS ops are in order (ISA: DS in-order) -> safe read
                const int r = (kb >> 3) + half;     // chunk covers 2 routes
                const float* xp = xrow + (size_t)r * 8;
                float d = 0.0f;
                for (int i = 0; i < 8; ++i)
                    d += xp[i] * uslot[bl * TSTRIDE + half * 8 + i];
                l_logits[bl * RSTRIDE + r] += d;    // unique (b,r) per lane
            }
        }
        __syncthreads();
    }
}

extern "C" void kernel_launch(void* const* d_in, const int* in_sizes, int n_in,
                              void* d_out, int out_size, void* d_ws, size_t ws_size,
                              hipStream_t stream) {
    const float* x = (const float*)d_in[0];        // (256, 1152, 8) fp32
    const float* w = (const float*)d_in[1];        // (10, 1152, 8, 16) fp32
    float* out = (float*)d_out;                    // (256, 10, 16) fp32
    dim3 grid(16, 10);                             // 16 batch tiles x 10 capsules
    capsule_route_kernel<<<grid, NTHREADS, 0, stream>>>(x, w, out);
}